// MessagePassingConvolution_22505628631554
// MI455X (gfx1250) — compile-verified
//
#include <hip/hip_runtime.h>

#define N_NODES    8192
#define N_NEIGH    16
#define MUL        128
#define RADIAL_DIM 64
#define NUM_IRREPS 640   // 2*MUL + 3*MUL
#define FEAT_DIM   512   // MUL + 3*MUL
#define MSG_DIM    1408  // 2*MUL + 3*MUL*3

typedef float v2f __attribute__((ext_vector_type(2)));
typedef float v8f __attribute__((ext_vector_type(8)));

// Hardware float atomic-add to global memory (no CAS loop).
__device__ __forceinline__ void gatomic_add_f32(float* p, float v) {
    asm volatile("global_atomic_add_f32 %0, %1, off"
                 :
                 : "v"((unsigned long long)p), "v"(v)
                 : "memory");
}

__global__ __launch_bounds__(256) void mpconv_kernel(
    const float* __restrict__ node_feats,
    const float* __restrict__ vectors,
    const float* __restrict__ radial_embedding,
    const int*   __restrict__ receivers,
    const float* __restrict__ W_radial,
    const float* __restrict__ b_radial,
    float* __restrict__ out)
{
    __shared__ float s_scal[MUL];                       // 0.5 KB
    __shared__ float s_vec[MUL * 3];                    // 1.5 KB
    __shared__ float s_n[N_NEIGH][4];                   // normalized edge vectors
    __shared__ int   s_recv[N_NEIGH];
    __shared__ float s_A[N_NEIGH][RADIAL_DIM];          // 4 KB radial-embedding tile
    __shared__ float s_radial[N_NEIGH][NUM_IRREPS];     // 40 KB GEMM result

    const int node = blockIdx.x;
    const int tid  = threadIdx.x;
    const int lane = tid & 31;
    const int wave = tid >> 5;

    // hint: W_radial (160 KB) is hot across all 8192 blocks -> keep in L2
    __builtin_prefetch(W_radial, 0, 3);

    // ---- stage node-local data into LDS ----------------------------------
    const float* nf = node_feats + (size_t)node * FEAT_DIM;
    for (int i = tid; i < FEAT_DIM; i += 256) {
        float v = nf[i];
        if (i < MUL) s_scal[i] = v;
        else         s_vec[i - MUL] = v;
    }
    const float* re = radial_embedding + (size_t)node * (N_NEIGH * RADIAL_DIM);
    for (int i = tid; i < N_NEIGH * RADIAL_DIM; i += 256)
        s_A[i >> 6][i & 63] = re[i];

    if (tid < N_NEIGH) {
        const float* vp = vectors + ((size_t)node * N_NEIGH + tid) * 3;
        float x = vp[0], y = vp[1], z = vp[2];
        float inv = rsqrtf(x * x + y * y + z * z);
        s_n[tid][0] = x * inv;
        s_n[tid][1] = y * inv;
        s_n[tid][2] = z * inv;
        s_recv[tid] = receivers[(size_t)node * N_NEIGH + tid];
    }
    __syncthreads();

    // ---- radial GEMM: (16 edges x 64) @ (64 x 640) + bias, fp32 WMMA -----
    // A 16x4 fragment: lanes 0-15 hold K = 0,1 ; lanes 16-31 hold K = 2,3
    // B 4x16 fragment: lanes 0-15 hold rows K = 0,1 ; lanes 16-31 rows 2,3
    // C/D 16x16:      VGPR v -> row v + 8*(lane>=16), col = lane & 15
    const int m      = lane & 15;          // edge row / output column index
    const int khalf  = (lane >> 4) * 2;    // 0 or 2
    const int rowbase = (lane >> 4) * 8;

    for (int t = wave; t < NUM_IRREPS / 16; t += 8) {
        const int n0 = t * 16;
        const float bias = b_radial[n0 + m];
        v8f acc;
#pragma unroll
        for (int v = 0; v < 8; ++v) acc[v] = bias;

#pragma unroll
        for (int s = 0; s < RADIAL_DIM / 4; ++s) {
            const int kb = 4 * s + khalf;
            v2f a, b;
            a[0] = s_A[m][kb];
            a[1] = s_A[m][kb + 1];
            b[0] = W_radial[(size_t)kb * NUM_IRREPS + n0 + m];
            b[1] = W_radial[(size_t)(kb + 1) * NUM_IRREPS + n0 + m];
            acc = __builtin_amdgcn_wmma_f32_16x16x4_f32(
                /*neg_a=*/false, a, /*neg_b=*/false, b,
                /*c_mod=*/(short)0, acc, /*reuse_a=*/false, /*reuse_b=*/false);
        }
#pragma unroll
        for (int v = 0; v < 8; ++v)
            s_radial[rowbase + v][n0 + m] = acc[v];
    }
    __syncthreads();

    // ---- message construction + scatter-add ------------------------------
    // messages[k] (1408) = [ s*r, (v.n)*r, v_i*r, s*sqrt3*n_i*r, K12*(dot*n_i - v_i/3)*r ]
    const float SQRT3 = 1.7320508075688772f;
    const float K12   = 1.9364916731037085f;  // sqrt(15)/2

#pragma unroll
    for (int it = 0; it < (N_NEIGH * MUL) / 256; ++it) {
        const int idx = it * 256 + tid;
        const int k = idx >> 7;     // edge
        const int c = idx & 127;    // channel

        const float s  = s_scal[c];
        const float v0 = s_vec[3 * c + 0];
        const float v1 = s_vec[3 * c + 1];
        const float v2 = s_vec[3 * c + 2];
        const float nx = s_n[k][0], ny = s_n[k][1], nz = s_n[k][2];
        const float dot = v0 * nx + v1 * ny + v2 * nz;

        const float r0  = s_radial[k][c];
        const float r1  = s_radial[k][MUL + c];
        const float rv0 = s_radial[k][2 * MUL + c];
        const float rv1 = s_radial[k][3 * MUL + c];
        const float rv2 = s_radial[k][4 * MUL + c];

        float* orow = out + (size_t)s_recv[k] * MSG_DIM;

        // scalar messages
        gatomic_add_f32(orow + c,        s   * r0);
        gatomic_add_f32(orow + MUL + c,  dot * r1);

        // vec broadcast (m = c)
        gatomic_add_f32(orow + 2 * MUL + 3 * c + 0, v0 * rv0);
        gatomic_add_f32(orow + 2 * MUL + 3 * c + 1, v1 * rv0);
        gatomic_add_f32(orow + 2 * MUL + 3 * c + 2, v2 * rv0);

        // tp_1o_a (m = MUL + c): scal * y1
        const float sa = s * SQRT3 * rv1;
        gatomic_add_f32(orow + 2 * MUL + 3 * (MUL + c) + 0, sa * nx);
        gatomic_add_f32(orow + 2 * MUL + 3 * (MUL + c) + 1, sa * ny);
        gatomic_add_f32(orow + 2 * MUL + 3 * (MUL + c) + 2, sa * nz);

        // tp_1o_b (m = 2*MUL + c): K12 * (T @ v), T = n n^T - I/3
        const float kb = K12 * rv2;
        gatomic_add_f32(orow + 2 * MUL + 3 * (2 * MUL + c) + 0, kb * (dot * nx - v0 * (1.0f / 3.0f)));
        gatomic_add_f32(orow + 2 * MUL + 3 * (2 * MUL + c) + 1, kb * (dot * ny - v1 * (1.0f / 3.0f)));
        gatomic_add_f32(orow + 2 * MUL + 3 * (2 * MUL + c) + 2, kb * (dot * nz - v2 * (1.0f / 3.0f)));
    }
}

extern "C" void kernel_launch(void* const* d_in, const int* in_sizes, int n_in,
                              void* d_out, int out_size, void* d_ws, size_t ws_size,
                              hipStream_t stream) {
    (void)in_sizes; (void)n_in; (void)d_ws; (void)ws_size;

    const float* node_feats       = (const float*)d_in[0];
    const float* vectors          = (const float*)d_in[1];
    const float* radial_embedding = (const float*)d_in[2];
    const int*   receivers        = (const int*)d_in[3];
    const float* W_radial         = (const float*)d_in[4];
    const float* b_radial         = (const float*)d_in[5];
    float* out = (float*)d_out;

    // output is accumulated with atomics -> must start from zero each call
    hipMemsetAsync(out, 0, (size_t)out_size * sizeof(float), stream);

    mpconv_kernel<<<N_NODES, 256, 0, stream>>>(
        node_feats, vectors, radial_embedding, receivers, W_radial, b_radial, out);
}